// NormalPolicy_60696477827605
// MI455X (gfx1250) — compile-verified
//
#include <hip/hip_runtime.h>
#include <math.h>

typedef __attribute__((ext_vector_type(2))) float v2f;
typedef __attribute__((ext_vector_type(8))) float v8f;

#define BB 8
#define NN 500000
#define SS 4096
#define NV 300
#define NVPAD 304
#define SEEDS_PER_BLOCK 128

__global__ __launch_bounds__(256) void grasp_view_wmma_kernel(
    const float* __restrict__ cloud_normals,   // [B, N, 3]
    const int*   __restrict__ idxs,            // [B, S]
    float*       __restrict__ out)             // [B*S] inds | [B*S*3] xyz | [B*S*9] rot
{
    __shared__ float4 sviews[NVPAD];           // {x, y, z, |v|^2}
    const int tid = threadIdx.x;

    // --- stage Fibonacci-sphere view codebook (double math, cast to f32) ---
    for (int v = tid; v < NVPAD; v += 256) {
        if (v < NV) {
            const double PHI = 0.61803398874989484820;  // (sqrt(5)-1)/2
            const double PI_ = 3.14159265358979323846;
            double zi = (2.0 * (double)v + 1.0) / (double)NV - 1.0;
            double rr = sqrt(fmax(1.0 - zi * zi, 0.0));
            double ang = 2.0 * (double)v * PI_ * PHI;
            float x = (float)(rr * cos(ang));
            float y = (float)(rr * sin(ang));
            float z = (float)zi;
            sviews[v] = make_float4(x, y, z, x * x + y * y + z * z);
        } else {
            sviews[v] = make_float4(0.f, 0.f, 0.f, __builtin_inff());
        }
    }
    __syncthreads();

    const int wave      = tid >> 5;
    const int lane      = tid & 31;
    const int m         = lane & 15;       // seed row within wave tile
    const int khalf     = lane >> 4;       // 0: K=0,1   1: K=2,3
    const int seed_base = blockIdx.x * SEEDS_PER_BLOCK + wave * 16;
    const int g         = seed_base + m;   // flat seed id in [0, B*S)
    const int b         = g >> 12;         // S = 4096

    // --- gather seed normal (random L2-resident access) ---
    const int idx = idxs[g];
    const long long nbase = ((long long)b * NN + (long long)idx) * 3;
    const float vx = cloud_normals[nbase + 0];
    const float vy = cloud_normals[nbase + 1];
    const float vz = cloud_normals[nbase + 2];

    // --- A matrix 16x4 f32: (x,y | z,0) per ISA layout ---
    v2f a;
    a.x = khalf ? vz : vx;
    a.y = khalf ? 0.f : vy;

    float bestd[8];
    int   bestn[8];
#pragma unroll
    for (int r = 0; r < 8; ++r) { bestd[r] = __builtin_inff(); bestn[r] = 0; }

    // --- 19 tiles of V_WMMA_F32_16X16X4_F32 over the 304-padded codebook ---
    // Per-lane candidate index n is strictly increasing, so a strict '<'
    // keeps the first (lowest-index) minimum: no in-loop tie-break needed.
#pragma unroll 1
    for (int n0 = 0; n0 < NVPAD; n0 += 16) {
        const int n = n0 + m;
        float4 vw = sviews[n];
        v2f bm;
        bm.x = khalf ? vw.z : vw.x;
        bm.y = khalf ? 0.f : vw.y;
        v8f c = {};
        c = __builtin_amdgcn_wmma_f32_16x16x4_f32(
                /*neg_a=*/false, a, /*neg_b=*/false, bm,
                /*c_mod=*/(short)0, c, /*reuse_a=*/false, /*reuse_b=*/false);
#pragma unroll
        for (int r = 0; r < 8; ++r) {
            float d2 = fmaf(-2.0f, c[r], vw.w);  // ||v||^2 - 2 x.v
            if (d2 < bestd[r]) { bestd[r] = d2; bestn[r] = n; }
        }
    }

    // --- argmin reduction across the 16 lanes of each half-wave ---
    // (xor masks 1..8 keep the two 16-lane halves independent in wave32)
#pragma unroll
    for (int r = 0; r < 8; ++r) {
#pragma unroll
        for (int mk = 1; mk < 16; mk <<= 1) {
            float od = __shfl_xor(bestd[r], mk, 32);
            int   on = __shfl_xor(bestn[r], mk, 32);
            if (od < bestd[r] || (od == bestd[r] && on < bestn[r])) {
                bestd[r] = od; bestn[r] = on;
            }
        }
    }

    const int OFF0 = 0;
    const int OFF1 = BB * SS;        // vp_xyz
    const int OFF2 = BB * SS * 4;    // vp_rot

    // lane 0 holds rows M=0..7, lane 16 holds rows M=8..15
    if (lane == 0 || lane == 16) {
#pragma unroll
        for (int r = 0; r < 8; ++r) {
            int seed = seed_base + khalf * 8 + r;
            out[OFF0 + seed] = (float)bestn[r];
        }
    }

    if (khalf == 0) {
        // vp_xyz
        out[OFF1 + (long long)g * 3 + 0] = vx;
        out[OFF1 + (long long)g * 3 + 1] = vy;
        out[OFF1 + (long long)g * 3 + 2] = vz;

        // rotation frame from towards = -normal, angle = 0 -> R = [ax|ay|az]
        float axx = -vx, axy = -vy, axz = -vz;
        float ayx = vy, ayy = -vx, ayz = 0.f;       // [-ax.y, ax.x, 0]
        float n2 = ayx * ayx + ayy * ayy;
        if (n2 == 0.f) { ayx = 0.f; ayy = 1.f; ayz = 0.f; n2 = 1.f; }
        float inva = 1.0f / sqrtf(axx * axx + axy * axy + axz * axz);
        float invy = 1.0f / sqrtf(n2);
        float anx = axx * inva, any_ = axy * inva, anz = axz * inva;
        float bnx = ayx * invy, bny = ayy * invy, bnz = ayz * invy;
        float czx = any_ * bnz - anz * bny;
        float czy = anz * bnx - anx * bnz;
        float czz = anx * bny - any_ * bnx;
        float* R = out + OFF2 + (long long)g * 9;
        R[0] = anx;  R[1] = bnx;  R[2] = czx;
        R[3] = any_; R[4] = bny;  R[5] = czy;
        R[6] = anz;  R[7] = bnz;  R[8] = czz;
    }
}

extern "C" void kernel_launch(void* const* d_in, const int* in_sizes, int n_in,
                              void* d_out, int out_size, void* d_ws, size_t ws_size,
                              hipStream_t stream) {
    const float* cloud = (const float*)d_in[0];   // [8, 500000, 3] f32
    const int*   idxs  = (const int*)d_in[1];     // [8, 4096] i32
    float* out = (float*)d_out;                   // 425984 floats
    (void)in_sizes; (void)n_in; (void)out_size; (void)d_ws; (void)ws_size;
    dim3 grid((BB * SS) / SEEDS_PER_BLOCK);       // 256 blocks
    grasp_view_wmma_kernel<<<grid, 256, 0, stream>>>(cloud, idxs, out);
}